// SWE_Pooling_45896020525941
// MI455X (gfx1250) — compile-verified
//
#include <hip/hip_runtime.h>
#include <math.h>

// Problem sizes (fixed by the reference)
#define BB 8
#define NN 4096
#define DD 512
#define SS 256
#define MM 1024
#define EPSF 1.1920929e-07f  // float32 eps

typedef __attribute__((ext_vector_type(2))) float v2f;
typedef __attribute__((ext_vector_type(4))) float v4f;
typedef __attribute__((ext_vector_type(8))) float v8f;

// ---------------------------------------------------------------------------
// Kernel 1: W = v / ||v||  (row-wise, S rows of D floats)
// ---------------------------------------------------------------------------
__global__ __launch_bounds__(256) void k_normalize_w(const float* __restrict__ v,
                                                     float* __restrict__ W) {
    __shared__ float red[256];
    const int row = blockIdx.x;                       // 0..S-1
    const float* src = v + (size_t)row * DD;
    const float a0 = src[threadIdx.x];
    const float a1 = src[threadIdx.x + 256];
    red[threadIdx.x] = a0 * a0 + a1 * a1;
    __syncthreads();
    for (int off = 128; off > 0; off >>= 1) {
        if (threadIdx.x < off) red[threadIdx.x] += red[threadIdx.x + off];
        __syncthreads();
    }
    const float nrm = sqrtf(red[0]);
    float* dst = W + (size_t)row * DD;
    dst[threadIdx.x]       = a0 / nrm;
    dst[threadIdx.x + 256] = a1 / nrm;
}

// ---------------------------------------------------------------------------
// Kernel 2: refmean[s] = mean_m reference_pts[m, s]
// ---------------------------------------------------------------------------
__global__ __launch_bounds__(256) void k_refmean(const float* __restrict__ ref,
                                                 float* __restrict__ refmean) {
    const int s = threadIdx.x;                        // single block of 256
    float acc = 0.0f;
    for (int m = 0; m < MM; ++m) acc += ref[(size_t)m * SS + s];
    refmean[s] = acc * (1.0f / (float)MM);
}

// ---------------------------------------------------------------------------
// Kernel 3: coefficient vector c[n] such that mean_m(interp(y)) = sum_n c[n]*y_(n)
// Replicates searchsorted-left on x = linspace(0,1,N+2)[1:-1] and the EPS slope.
// ---------------------------------------------------------------------------
__global__ __launch_bounds__(1024) void k_coef(float* __restrict__ coef) {
    __shared__ float c[NN];
    for (int i = threadIdx.x; i < NN; i += 1024) c[i] = 0.0f;
    __syncthreads();

    const int m = threadIdx.x;                        // one thread per target point
    const float xnew = (float)(m + 1) / (float)(MM + 1);
    // searchsorted(x, xnew, 'left'): smallest i with x[i] >= xnew ; x[i]=(i+1)/(N+1)
    int lo = 0, hi = NN;
    while (lo < hi) {
        const int mid = (lo + hi) >> 1;
        const float xv = (float)(mid + 1) / (float)(NN + 1);
        if (xv < xnew) lo = mid + 1; else hi = mid;
    }
    int ind = lo - 1;
    if (ind < 0) ind = 0;
    if (ind > NN - 2) ind = NN - 2;
    const float x0 = (float)(ind + 1) / (float)(NN + 1);
    const float x1 = (float)(ind + 2) / (float)(NN + 1);
    const float w  = (xnew - x0) / (EPSF + (x1 - x0));   // ynew = y0 + (y1-y0)*w
    atomicAdd(&c[ind],     (1.0f - w) * (1.0f / (float)MM));   // ds_add_f32
    atomicAdd(&c[ind + 1],  w         * (1.0f / (float)MM));
    __syncthreads();
    for (int i = threadIdx.x; i < NN; i += 1024) coef[i] = c[i];
}

// ---------------------------------------------------------------------------
// Kernel 4: Xslices[b,n,s] = sum_d X[b,n,d] * W[s,d]  via V_WMMA_F32_16X16X4_F32
// One wave computes a 16(n) x 16(s) fp32 tile, K stepped by 4 over D=512.
// A-frag layout (ISA 16x4 f32): lane l<16 holds A[M=l][K=0..1], lane l+16 K=2..3
//   -> per-lane float2 from X row (contiguous in d).
// B-frag symmetric: lane l holds B[K][N=l] pairs = W[s0+l][k..k+1] (contiguous).
// D layout: VGPR r -> row M = r + 8*(lane>=16), col N = lane&15.
// ---------------------------------------------------------------------------
__global__ __launch_bounds__(256) void k_gemm_wmma(const float* __restrict__ X,
                                                   const float* __restrict__ W,
                                                   float* __restrict__ Y) {
    const int lane = threadIdx.x & 31;
    const int wave = threadIdx.x >> 5;                 // 8 waves per block
    const int l    = lane & 15;
    const int half = lane >> 4;                        // 0: K=0..1, 1: K=2..3
    const int b    = blockIdx.z;
    const int n0   = blockIdx.x * 128 + wave * 16;
    const int s0   = blockIdx.y * 16;

    const float* aptr = X + ((size_t)b * NN + (n0 + l)) * DD + 2 * half;
    const float* bptr = W + (size_t)(s0 + l) * DD + 2 * half;

    v8f c0 = {}; v8f c1 = {};
    #pragma unroll 2
    for (int k = 0; k < DD; k += 8) {
        // two K-steps per pointer, fetched as one b128 each
        const v4f av = *(const v4f*)aptr;              // X[.., k+2h .. k+2h+1], [k+2h+2 .. k+2h+3]
        const v4f bv = *(const v4f*)bptr;
        const v2f a0 = {av.x, av.y}, a1 = {av.z, av.w};
        const v2f b0 = {bv.x, bv.y}, b1 = {bv.z, bv.w};
        c0 = __builtin_amdgcn_wmma_f32_16x16x4_f32(false, a0, false, b0,
                                                   (short)0, c0, false, false);
        c1 = __builtin_amdgcn_wmma_f32_16x16x4_f32(false, a1, false, b1,
                                                   (short)0, c1, false, false);
        aptr += 8; bptr += 8;
    }
    const v8f c = c0 + c1;

    // Store row-major (B,N,S): lane-contiguous in s -> coalesced per half-wave
    float* out = Y + ((size_t)b * NN + (n0 + 8 * half)) * SS + (s0 + l);
    #pragma unroll
    for (int r = 0; r < 8; ++r) out[(size_t)r * SS] = c[r];
}

// ---------------------------------------------------------------------------
// Kernel 5: per (b,s): LDS bitonic sort of the 4096-column, then
// out[b*S+s] = refmean[s] - sum_n coef[n] * sorted[n]
// (gather-by-argsort(ref) is a permutation along the averaged axis -> no-op)
// ---------------------------------------------------------------------------
__global__ __launch_bounds__(1024) void k_sort_reduce(const float* __restrict__ Y,
                                                      const float* __restrict__ coef,
                                                      const float* __restrict__ refmean,
                                                      float* __restrict__ out) {
    __shared__ float sd[NN];
    __shared__ float red[1024];
    const int bs = blockIdx.x;
    const int b  = bs >> 8;
    const int s  = bs & 255;

    const float* col = Y + (size_t)b * NN * SS + s;
    for (int i = threadIdx.x; i < NN; i += 1024) sd[i] = col[(size_t)i * SS];
    __syncthreads();

    // ascending bitonic sort of 4096 elements
    for (int k = 2; k <= NN; k <<= 1) {
        for (int j = k >> 1; j > 0; j >>= 1) {
            for (int i = threadIdx.x; i < NN; i += 1024) {
                const int ixj = i ^ j;
                if (ixj > i) {
                    const float a  = sd[i];
                    const float bb = sd[ixj];
                    const bool up  = ((i & k) == 0);
                    if ((a > bb) == up) { sd[i] = bb; sd[ixj] = a; }
                }
            }
            __syncthreads();
        }
    }

    float partial = 0.0f;
    for (int i = threadIdx.x; i < NN; i += 1024) partial += coef[i] * sd[i];
    red[threadIdx.x] = partial;
    __syncthreads();
    for (int off = 512; off > 0; off >>= 1) {
        if (threadIdx.x < off) red[threadIdx.x] += red[threadIdx.x + off];
        __syncthreads();
    }
    if (threadIdx.x == 0) out[bs] = refmean[s] - red[0];
}

// ---------------------------------------------------------------------------
extern "C" void kernel_launch(void* const* d_in, const int* in_sizes, int n_in,
                              void* d_out, int out_size, void* d_ws, size_t ws_size,
                              hipStream_t stream) {
    (void)in_sizes; (void)n_in; (void)out_size; (void)ws_size;
    const float* X   = (const float*)d_in[0];   // (B,N,D)
    const float* v   = (const float*)d_in[1];   // (S,D)
    const float* ref = (const float*)d_in[2];   // (M,S)
    float* out = (float*)d_out;                 // (B,S)

    // workspace layout (floats)
    float* ws      = (float*)d_ws;
    float* Wn      = ws;                         // S*D      = 131072
    float* refmean = Wn + (size_t)SS * DD;       // 256
    float* coef    = refmean + SS;               // 4096
    float* Y       = coef + NN;                  // B*N*S    = 8388608  (~34 MB total)

    k_normalize_w<<<SS, 256, 0, stream>>>(v, Wn);
    k_refmean<<<1, 256, 0, stream>>>(ref, refmean);
    k_coef<<<1, 1024, 0, stream>>>(coef);
    k_gemm_wmma<<<dim3(NN / 128, SS / 16, BB), 256, 0, stream>>>(X, Wn, Y);
    k_sort_reduce<<<BB * SS, 1024, 0, stream>>>(Y, coef, refmean, out);
}